// GIOULOSS_19524921327670
// MI455X (gfx1250) — compile-verified
//
#include <hip/hip_runtime.h>
#include <math.h>

typedef __attribute__((ext_vector_type(2))) float v2f;
typedef __attribute__((ext_vector_type(8))) float v8f;

#define EPS_F   1e-7f
#define ACLIP_F (1.0f - 1e-6f)
#define PI_F    3.14159265358979323846f

// Fast ~1ulp helpers: v_rcp_f32 / v_sqrt_f32 without OCML refinement tails.
__device__ __forceinline__ float fdiv_fast(float a, float b) {
    return a * __builtin_amdgcn_rcpf(b);
}
__device__ __forceinline__ float fsqrt_fast(float a) {
    return __builtin_amdgcn_sqrtf(a);
}

// Exchange with lane^16 via one ds_swizzle (SWAPX16: xor=0x10, and=0x1F).
__device__ __forceinline__ float lane_xor16(float v) {
    return __int_as_float(__builtin_amdgcn_ds_swizzle(__float_as_int(v), 0x401F));
}

// ---------------------------------------------------------------------------
// Wave32 all-reduce using V_WMMA_F32_16X16X4_F32.
// A (16x4 f32, 2 VGPRs): VGPR0 = per-lane partial, VGPR1 = 0.
// B (4x16 f32, 2 VGPRs): all ones.
// D[m][n] = sum_k A[m][k].  Summing a lane's 8 D registers gives the sum of
// that lane-half's 8 rows of A; adding lane^16's value gives the sum of the
// entire A matrix = sum of all 32 lane partials (layout-robust: B uniform,
// both halves folded).  Requires EXEC == all ones at the call site.
// ---------------------------------------------------------------------------
__device__ __forceinline__ float wave_sum32(float v) {
    v2f a; a.x = v;    a.y = 0.0f;
    v2f b; b.x = 1.0f; b.y = 1.0f;
    v8f c = {};
    v8f d = __builtin_amdgcn_wmma_f32_16x16x4_f32(
        /*neg_a=*/false, a, /*neg_b=*/false, b,
        /*c_mod=*/(short)0, c, /*reuse_a=*/false, /*reuse_b=*/false);
    float s = ((d[0] + d[1]) + (d[2] + d[3])) + ((d[4] + d[5]) + (d[6] + d[7]));
    s += lane_xor16(s);
    return s;  // total of all 32 lanes, in every lane
}

// Block-wide sum (result valid in thread 0). blockDim.x <= 1024.
__device__ __forceinline__ float block_sum(float v) {
    __shared__ float lsum[32];
    float w   = wave_sum32(v);
    int lane  = threadIdx.x & 31;
    int wid   = threadIdx.x >> 5;
    if (lane == 0) lsum[wid] = w;
    __syncthreads();
    float r = 0.0f;
    if (threadIdx.x == 0) {
        int nw = (blockDim.x + 31) >> 5;
        for (int i = 0; i < nw; ++i) r += lsum[i];
    }
    return r;
}

// ---------------------------------------------------------------------------
// Circle GIoU loss for one pair (fp32, mirrors the JAX reference).
// ---------------------------------------------------------------------------
__device__ __forceinline__ float circle_giou(float cx0, float cy0, float r0,
                                             float cx1, float cy1, float r1) {
    float dx = cx0 - cx1, dy = cy0 - cy1;
    float d2 = dx * dx + dy * dy;
    float d  = fsqrt_fast(fmaxf(d2, EPS_F));
    float rmax = fmaxf(r0, r1), rmin = fminf(r0, r1);
    float rsum = r0 + r1, rdif = rmax - rmin;
    bool lens      = (d < rsum) && (d > rdif);
    bool contained = (d <= rdif);
    float r0s = r0 * r0, r1s = r1 * r1;

    float cos0 = fminf(fmaxf(fdiv_fast(d2 + r0s - r1s, fmaxf(2.0f * d * r0, EPS_F)), -ACLIP_F), ACLIP_F);
    float cos1 = fminf(fmaxf(fdiv_fast(d2 + r1s - r0s, fmaxf(2.0f * d * r1, EPS_F)), -ACLIP_F), ACLIP_F);

    float t = (rsum - d) * (d + r0 - r1) * (d - r0 + r1) * (d + rsum);
    float t_safe = lens ? fmaxf(t, EPS_F) : 1.0f;
    float lens_area = r0s * acosf(cos0) + r1s * acosf(cos1) - 0.5f * fsqrt_fast(t_safe);

    float inter = lens ? lens_area : (contained ? PI_F * rmin * rmin : 0.0f);
    float uni   = PI_F * (r0s + r1s) - inter;
    float iou   = fdiv_fast(inter, fmaxf(uni, EPS_F));

    float alpha   = acosf(fminf(fmaxf(fdiv_fast(rdif, d), 0.0f), ACLIP_F));
    float h2      = d2 - rdif * rdif;
    float h2_safe = contained ? 1.0f : fmaxf(h2, EPS_F);
    float hull_open = rmax * rmax * (PI_F - alpha) + rmin * rmin * alpha
                    + (rmax + rmin) * fsqrt_fast(h2_safe);
    float hull = contained ? PI_F * rmax * rmax : hull_open;

    return 1.0f - (iou - fdiv_fast(hull - uni, fmaxf(hull, EPS_F)));
}

// ---------------------------------------------------------------------------
// Kernel 1: 4 rows/thread via 3 aligned float4 loads per input (pure b128
// traffic, 100% utilized), per-block WMMA reduction -> d_ws[block].
// ---------------------------------------------------------------------------
__global__ void __launch_bounds__(256)
giou_partial_kernel(const float* __restrict__ x, const float* __restrict__ y,
                    float* __restrict__ partials, int nRows) {
    const int  tid  = blockIdx.x * blockDim.x + threadIdx.x;
    const long base = (long)tid * 4;
    float acc = 0.0f;

    if (base + 3 < (long)nRows) {
        const float4* xv = (const float4*)x + (long)tid * 3;
        const float4* yv = (const float4*)y + (long)tid * 3;
        float4 x0 = xv[0], x1 = xv[1], x2 = xv[2];
        float4 y0 = yv[0], y1 = yv[1], y2 = yv[2];
        acc += circle_giou(x0.x, x0.y, x0.z,  y0.x, y0.y, y0.z);
        acc += circle_giou(x0.w, x1.x, x1.y,  y0.w, y1.x, y1.y);
        acc += circle_giou(x1.z, x1.w, x2.x,  y1.z, y1.w, y2.x);
        acc += circle_giou(x2.y, x2.z, x2.w,  y2.y, y2.z, y2.w);
    } else {
        for (long i = base; i < (long)nRows; ++i) {
            acc += circle_giou(x[i*3], x[i*3+1], x[i*3+2],
                               y[i*3], y[i*3+1], y[i*3+2]);
        }
    }
    // Reconverged here: EXEC is all ones for the WMMA reduce.
    float bsum = block_sum(acc);
    if (threadIdx.x == 0) partials[blockIdx.x] = bsum;
}

// ---------------------------------------------------------------------------
// Kernel 2: single block reduces all block partials -> d_out[0].
// ---------------------------------------------------------------------------
__global__ void __launch_bounds__(256)
reduce_partials_kernel(const float* __restrict__ partials, int n,
                       float* __restrict__ out) {
    float acc = 0.0f;
    for (int i = threadIdx.x; i < n; i += blockDim.x) acc += partials[i];
    float s = block_sum(acc);
    if (threadIdx.x == 0) out[0] = s;
}

extern "C" void kernel_launch(void* const* d_in, const int* in_sizes, int n_in,
                              void* d_out, int out_size, void* d_ws, size_t ws_size,
                              hipStream_t stream) {
    const float* x = (const float*)d_in[0];
    const float* y = (const float*)d_in[1];
    float* out      = (float*)d_out;
    float* partials = (float*)d_ws;

    const int nRows        = in_sizes[0] / 3;      // (N,3) fp32
    const int rowsPerBlock = 256 * 4;
    const int nBlocks      = (nRows + rowsPerBlock - 1) / rowsPerBlock;

    giou_partial_kernel<<<nBlocks, 256, 0, stream>>>(x, y, partials, nRows);
    reduce_partials_kernel<<<1, 256, 0, stream>>>(partials, nBlocks, out);
}